// RandomPunisher_62680752718004
// MI455X (gfx1250) — compile-verified
//
#include <hip/hip_runtime.h>
#include <stdint.h>

// Connect-4 "winning move" kernel for MI455X (gfx1250, wave32).
// One board (6x7 floats in {-1,0,1}) per thread. Bitboard layout:
// bit = col*8 + row (rows 0..5 real, rows 6..7 sentinel zeros), so the four
// line directions are bit-shifts by 1 (vert), 8 (horiz), 9 (diag), 7 (anti).

#define C4_ROWS 6
#define C4_COLS 7
#define FPB 42                    // floats per board
#define TB 256                    // threads (= boards) per block, 8 waves
#define TILE_F (TB * FPB)         // 10752 floats staged per block (43008 B)
#define TILE_F4 (TILE_F / 4)      // 2688 float4 chunks (10.5 * 256)
#define OUT_F (TB * C4_COLS)      // 1792 output floats per block (7168 B)
#define OUT_F4 (OUT_F / 4)        // 448 float4 chunks (1.75 * 256)

// ---- CDNA5 async global<->LDS path (ASYNCcnt-tracked, CDNA5 ISA) ----
#if defined(__HIP_DEVICE_COMPILE__) && defined(__gfx1250__)
#if __has_builtin(__builtin_amdgcn_global_load_async_to_lds_b128) && \
    __has_builtin(__builtin_amdgcn_global_store_async_from_lds_b32)
#define USE_ASYNC_LDS 1
#if __has_builtin(__builtin_amdgcn_global_store_async_from_lds_b128)
#define HAVE_ST128 1
#endif
#if __has_builtin(__builtin_amdgcn_s_wait_asynccnt)
#define ASYNC_WAIT0() __builtin_amdgcn_s_wait_asynccnt(0)
#else
#define ASYNC_WAIT0() asm volatile("s_wait_asynccnt 0x0" ::: "memory")
#endif
#endif
#endif

#ifdef USE_ASYNC_LDS
typedef int v4i __attribute__((ext_vector_type(4)));
typedef __attribute__((address_space(1))) v4i as1_v4i;
typedef __attribute__((address_space(3))) v4i as3_v4i;
typedef __attribute__((address_space(1))) int as1_int;
typedef __attribute__((address_space(3))) int as3_int;
typedef __attribute__((address_space(3))) unsigned char as3_byte;
// Generic LDS pointer -> AS3: low 32 bits of a flat LDS address are the LDS
// byte offset (ISA 10.2 aperture rules); AS3 pointers are 32-bit.
__device__ __forceinline__ as3_byte* to_lds(const void* p) {
  return (as3_byte*)(uint32_t)(uintptr_t)p;
}
// INST_OFFSET is added to BOTH the global and the LDS address (ISA 08 / 4.4),
// so one per-lane base serves every staged transfer via immediate offsets.
#define STAGE_IN(I)                                              \
  __builtin_amdgcn_global_load_async_to_lds_b128(                \
      (as1_v4i*)(uintptr_t)gin, (as3_v4i*)lin, (I) * 4096, 0)
#ifdef HAVE_ST128
#define STAGE_OUT(I)                                             \
  __builtin_amdgcn_global_store_async_from_lds_b128(             \
      (as1_v4i*)(uintptr_t)gout, (as3_v4i*)lout, (I) * 4096, 0)
#else
#define STAGE_OUT32(I)                                           \
  __builtin_amdgcn_global_store_async_from_lds_b32(              \
      (as1_int*)(uintptr_t)gout, (as3_int*)lout, (I) * 1024, 0)
#endif
#endif

// All cells e such that M | {e} contains 4-in-a-row through e.
__device__ __forceinline__ unsigned long long win_cells(unsigned long long m) {
  unsigned long long w = 0ull;
  const int dirs[4] = {1, 7, 8, 9};
#pragma unroll
  for (int i = 0; i < 4; ++i) {
    const int d = dirs[i];
    const unsigned long long r1 = m >> d, r2 = m >> (2 * d), r3 = m >> (3 * d);
    const unsigned long long l1 = m << d, l2 = m << (2 * d), l3 = m << (3 * d);
    const unsigned long long r12 = r1 & r2, l12 = l1 & l2;
    w |= (r12 & r3) | (r12 & l1) | (l12 & r1) | (l12 & l3);
  }
  return w;
}

__global__ __launch_bounds__(TB) void connect4_winmove_kernel(
    const float* __restrict__ x, float* __restrict__ out, int nBoards) {
  __shared__ __align__(16) unsigned char smem[TILE_F * 4 + OUT_F * 4];
  float* s_in = (float*)smem;
  float* s_out = (float*)(smem + TILE_F * 4);

  const int tid = threadIdx.x;
  const unsigned long long totalF = (unsigned long long)nBoards * FPB;
  const unsigned long long blockF = (unsigned long long)blockIdx.x * TILE_F;
  const bool fullTile = (blockF + TILE_F) <= totalF;  // true for all blocks here

  // ---- Stage the block's input tile into LDS, fully coalesced ----
  if (fullTile) {
#ifdef USE_ASYNC_LDS
    const float* gin = x + blockF + (size_t)tid * 4u;       // 16 B per lane
    as3_byte* lin = to_lds(s_in) + (size_t)tid * 16u;
    STAGE_IN(0); STAGE_IN(1); STAGE_IN(2); STAGE_IN(3); STAGE_IN(4);
    STAGE_IN(5); STAGE_IN(6); STAGE_IN(7); STAGE_IN(8); STAGE_IN(9);
    if (tid < TB / 2) STAGE_IN(10);                         // 2688 = 10.5*256
    ASYNC_WAIT0();
#else
#pragma unroll
    for (int i = 0; i < 11; ++i) {
      const unsigned idx = i * TB + tid;
      if (i < 10 || tid < TB / 2) {
        const float4 v = *(const float4*)(x + blockF + (size_t)idx * 4u);
        *(float4*)(s_in + (size_t)idx * 4u) = v;
      }
    }
#endif
  } else {  // ragged tail block: scalar, exact dword granularity
    for (int idx = tid; idx < TILE_F; idx += TB) {
      const unsigned long long g = blockF + idx;
      if (g < totalF) s_in[idx] = x[g];
    }
  }
  __syncthreads();

  // ---- Per-thread board evaluation from LDS (bank-conflict-free: 42 mod 64) ----
  {
    unsigned long long occ64 = 0ull, p1s = 0ull;
    const uint2* bp = (const uint2*)(s_in + (size_t)tid * FPB);  // 8B aligned
    auto acc = [&](unsigned u, int j) {
      // u in {0, 0x3F800000, 0xBF800000}: s = u>>29 -> bit0 = occ, bit2 = neg
      const int pos = (j % 7) * 8 + j / 7;
      const unsigned s = u >> 29;
      occ64 |= (unsigned long long)(s & 1u) << pos;
      p1s   |= (unsigned long long)(s & 4u) << pos;  // neg, pre-shifted by +2
    };
#pragma unroll
    for (int k = 0; k < FPB / 2; ++k) {
      const uint2 wv = bp[k];
      acc(wv.x, 2 * k);
      acc(wv.y, 2 * k + 1);
    }
    const unsigned long long p1 = p1s >> 2;   // -1 stones
    const unsigned long long p0 = occ64 ^ p1; // +1 stones

    const unsigned long long W0 = win_cells(p0);
    const unsigned long long W1 = win_cells(p1);

    unsigned m14 = 0;  // bits 0..6: +1 wins per col; bits 7..13: -1 wins
#pragma unroll
    for (int c = 0; c < C4_COLS; ++c) {
      const unsigned e = (~(unsigned)(occ64 >> (8 * c))) & 0x3Fu;  // empty rows
      const unsigned playable = e & 1u;                            // row 0 empty
      const int land = 31 - __clz((int)(e | 1u));                  // deepest empty
      const int pos = 8 * c + land;
      const unsigned w0 = ((unsigned)(W0 >> pos) & 1u) & playable;
      const unsigned w1 = ((unsigned)(W1 >> pos) & 1u) & playable;
      m14 |= (w0 << c) | (w1 << (c + C4_COLS));
    }

    int col = 0;
    if (m14) col = (__ffs((int)m14) - 1) % C4_COLS;
    // log(one_hot + 1e-12) in fp32: hot -> log(1.0f)=0.0f, cold -> log(1e-12f)
    const float LOGEPS = -27.631021f;
#pragma unroll
    for (int j = 0; j < C4_COLS; ++j) {
      float v = 0.0f;
      if (m14) v = (j == col) ? 0.0f : LOGEPS;
      s_out[(size_t)tid * C4_COLS + j] = v;
    }
  }
  __syncthreads();

  // ---- Coalesced output store (LDS -> global) ----
  const unsigned long long totalO = (unsigned long long)nBoards * C4_COLS;
  const unsigned long long blockO = (unsigned long long)blockIdx.x * OUT_F;
  if (fullTile) {
#ifdef USE_ASYNC_LDS
    float* gout = out + blockO + (size_t)tid * 4u;
    as3_byte* lout = to_lds(s_out) + (size_t)tid * 16u;
#ifdef HAVE_ST128
    STAGE_OUT(0);
    if (tid < 3 * TB / 4) STAGE_OUT(1);                     // 448 = 1.75*256
#else
    lout = to_lds(s_out) + (size_t)tid * 4u;
    gout = out + blockO + tid;
    STAGE_OUT32(0); STAGE_OUT32(1); STAGE_OUT32(2); STAGE_OUT32(3);
    STAGE_OUT32(4); STAGE_OUT32(5); STAGE_OUT32(6);
#endif
    ASYNC_WAIT0();
#else
#pragma unroll
    for (int i = 0; i < C4_COLS; ++i) {
      const unsigned idx = i * TB + tid;
      out[blockO + idx] = s_out[idx];
    }
#endif
  } else {
    for (int idx = tid; idx < OUT_F; idx += TB) {
      const unsigned long long g = blockO + idx;
      if (g < totalO) out[g] = s_out[idx];
    }
  }
}

extern "C" void kernel_launch(void* const* d_in, const int* in_sizes, int n_in,
                              void* d_out, int out_size, void* d_ws, size_t ws_size,
                              hipStream_t stream) {
  (void)n_in; (void)out_size; (void)d_ws; (void)ws_size;
  const float* x = (const float*)d_in[0];
  float* out = (float*)d_out;
  const int nBoards = in_sizes[0] / FPB;  // 524288
  const int grid = (nBoards + TB - 1) / TB;
  hipLaunchKernelGGL(connect4_winmove_kernel, dim3(grid), dim3(TB), 0, stream,
                     x, out, nBoards);
}